// SpatialGraphConv_49168785605185
// MI455X (gfx1250) — compile-verified
//
#include <hip/hip_runtime.h>

typedef __attribute__((ext_vector_type(2))) float v2f;
typedef __attribute__((ext_vector_type(8))) float v8f;

#define EPS_   1e-5f
#define NB     64
#define CIN    64
#define COUT   128
#define TDIM   300
#define VDIM   25
#define TT     4                 // time-steps per workgroup
#define PUSE   (TT*VDIM)         // 100 valid GEMM rows
#define MROWS  112               // padded to 7 * 16
#define MT     7
#define KDIM   128               // 64 z-channels + 64 x-channels
#define LDK    132               // padded K stride (dwords): banks (4p+k)%64 all distinct

// workspace layout (floats):
//  [0 .. 16384)   Wb: packed B operands, [kb=32][ntile=8][lane=32][2]
//  [16384..16512) bias (128)
//  [16512..17137) Aeff (625)

__global__ void prep_kernel(const float* __restrict__ A, const float* __restrict__ PA,
                            const float* __restrict__ conv_w, const float* __restrict__ conv_b,
                            const float* __restrict__ bn_g, const float* __restrict__ bn_b,
                            const float* __restrict__ bn_m, const float* __restrict__ bn_v,
                            const float* __restrict__ res_w, const float* __restrict__ res_b,
                            const float* __restrict__ res_g, const float* __restrict__ res_be,
                            const float* __restrict__ res_m, const float* __restrict__ res_v,
                            float* __restrict__ ws)
{
    int tid = blockIdx.x * blockDim.x + threadIdx.x;
    float* Wb   = ws;
    float* bias = ws + 16384;
    float* Aeff = ws + 16512;

    if (tid < 8192) {
        // tid = ((kb*8 + ntile)*32 + lane)
        int lane = tid & 31;
        int nt   = (tid >> 5) & 7;
        int kb   = tid >> 8;
        int o    = nt * 16 + (lane & 15);
        int kbase = kb * 4 + ((lane >> 4) << 1);   // lanes 0-15: K rows 0/1, lanes 16-31: 2/3
        float s1 = bn_g[o]  * rsqrtf(bn_v[o]  + EPS_);
        float s2 = res_g[o] * rsqrtf(res_v[o] + EPS_);
        int k0 = kbase, k1 = kbase + 1;
        float w0 = (k0 < CIN) ? conv_w[o*CIN + k0] * s1 : res_w[o*CIN + (k0-CIN)] * s2;
        float w1 = (k1 < CIN) ? conv_w[o*CIN + k1] * s1 : res_w[o*CIN + (k1-CIN)] * s2;
        Wb[tid*2 + 0] = w0;
        Wb[tid*2 + 1] = w1;
    } else if (tid < 8192 + 128) {
        int o = tid - 8192;
        float s1 = bn_g[o]  * rsqrtf(bn_v[o]  + EPS_);
        float s2 = res_g[o] * rsqrtf(res_v[o] + EPS_);
        bias[o] = conv_b[o]*s1 + bn_b[o]  - bn_m[o]*s1
                + res_b[o]*s2 + res_be[o] - res_m[o]*s2;
    } else if (tid < 8192 + 128 + VDIM*VDIM) {
        int i = tid - 8192 - 128;
        Aeff[i] = A[i] + PA[i];
    }
}

__global__ __launch_bounds__(256) void gcn_fused_kernel(const float* __restrict__ x,
                                                        const float* __restrict__ ws,
                                                        float* __restrict__ out)
{
    __shared__ float tile[MROWS * LDK];   // stage A: [p][ zcols 0..63 | xcols 64..127 ], stage B: [p][o]
    __shared__ float aeffS[VDIM * VDIM];
    __shared__ float biasS[COUT];

    const float* Wb    = ws;
    const float* biasG = ws + 16384;
    const float* aeffG = ws + 16512;

    const int tid = threadIdx.x;
    const int bid = blockIdx.x;
    const int n   = bid / (TDIM / TT);
    const int t0  = (bid % (TDIM / TT)) * TT;

    for (int i = tid; i < VDIM*VDIM; i += 256) aeffS[i] = aeffG[i];
    if (tid < COUT) biasS[tid] = biasG[tid];

    // ---- load x tile into K-columns [64..127]; zero the padded rows ----
    const float* xb = x + (size_t)n * CIN * TDIM * VDIM + (size_t)t0 * VDIM;
    for (int e = tid; e < CIN * PUSE; e += 256) {
        int c = e / PUSE;
        int r = e - c * PUSE;                       // r = tl*25 + v  (position row)
        float val = __builtin_nontemporal_load(&xb[(size_t)c * TDIM * VDIM + r]);
        tile[r * LDK + CIN + c] = val;
    }
    for (int e = tid; e < (MROWS - PUSE) * KDIM; e += 256) {
        int rr = PUSE + e / KDIM;
        int k  = e - (e / KDIM) * KDIM;
        tile[rr * LDK + k] = 0.0f;
    }
    __syncthreads();

    // ---- stage 1: graph aggregation  z[c,t,w] = sum_v x[c,t,v] * Aeff[v,w]  into cols [0..63]
    // (writes cols <64, reads cols >=64 -> disjoint, no barrier needed inside)
    for (int e = tid; e < CIN * PUSE; e += 256) {
        int c  = e / PUSE;
        int r  = e - c * PUSE;
        int tl = r / VDIM;
        int w  = r - tl * VDIM;
        const float* xr = &tile[(tl * VDIM) * LDK + CIN + c];
        float acc = 0.0f;
        #pragma unroll
        for (int v = 0; v < VDIM; ++v)
            acc += xr[v * LDK] * aeffS[v * VDIM + w];
        tile[r * LDK + c] = acc;
    }
    __syncthreads();

    // ---- stage 2: GEMM  out[p,o] = sum_k tile[p,k] * W[k,o]  via v_wmma_f32_16x16x4_f32
    const int lane = tid & 31;
    const int wv   = tid >> 5;                 // wave owns N-tile wv (16 output channels)
    const int mrow = lane & 15;                // A operand: lanes 0-15 rows, lanes 16-31 same rows K+2
    const int kh   = (lane >> 4) << 1;

    v8f acc[MT];
    #pragma unroll
    for (int i = 0; i < MT; ++i) acc[i] = (v8f){0.f,0.f,0.f,0.f,0.f,0.f,0.f,0.f};

    for (int kb = 0; kb < KDIM / 4; ++kb) {
        // B operand: pre-packed, fully coalesced b64 per wave
        v2f bv = *(const v2f*)(Wb + ((size_t)((kb * 8 + wv) * 32 + lane)) * 2);
        const int ko = kb * 4 + kh;
        #pragma unroll
        for (int mt = 0; mt < MT; ++mt) {
            v2f av = *(const v2f*)(&tile[(mt * 16 + mrow) * LDK + ko]);  // ds_load_b64, conflict-free
            acc[mt] = __builtin_amdgcn_wmma_f32_16x16x4_f32(
                false, av, false, bv, (short)0, acc[mt], false, false);
        }
    }
    __syncthreads();   // everyone done reading tile as A before we overwrite it

    // ---- transpose through LDS: D layout -> tile[p][o]
    const int ocol    = wv * 16 + (lane & 15);
    const int prow_hi = (lane >> 4) << 3;      // lanes 16-31 hold rows M+8
    #pragma unroll
    for (int mt = 0; mt < MT; ++mt) {
        #pragma unroll
        for (int j = 0; j < 8; ++j) {
            int p = mt * 16 + j + prow_hi;
            tile[p * LDK + ocol] = acc[mt][j];
        }
    }
    __syncthreads();

    // ---- fused bias + ReLU, contiguous 100-float runs per output channel ----
    float* outb = out + (size_t)n * COUT * TDIM * VDIM + (size_t)t0 * VDIM;
    for (int e = tid; e < COUT * PUSE; e += 256) {
        int o = e / PUSE;
        int r = e - o * PUSE;
        float val = tile[r * LDK + o] + biasS[o];
        val = fmaxf(val, 0.0f);
        __builtin_nontemporal_store(val, &outb[(size_t)o * TDIM * VDIM + r]);
    }
}

extern "C" void kernel_launch(void* const* d_in, const int* in_sizes, int n_in,
                              void* d_out, int out_size, void* d_ws, size_t ws_size,
                              hipStream_t stream)
{
    const float* x      = (const float*)d_in[0];
    const float* A      = (const float*)d_in[1];
    const float* PA     = (const float*)d_in[2];
    const float* conv_w = (const float*)d_in[3];
    const float* conv_b = (const float*)d_in[4];
    const float* bn_g   = (const float*)d_in[5];
    const float* bn_b   = (const float*)d_in[6];
    const float* bn_m   = (const float*)d_in[7];
    const float* bn_v   = (const float*)d_in[8];
    const float* res_w  = (const float*)d_in[9];
    const float* res_b  = (const float*)d_in[10];
    const float* res_g  = (const float*)d_in[11];
    const float* res_be = (const float*)d_in[12];
    const float* res_m  = (const float*)d_in[13];
    const float* res_v  = (const float*)d_in[14];
    float* out = (float*)d_out;
    float* ws  = (float*)d_ws;

    prep_kernel<<<35, 256, 0, stream>>>(A, PA, conv_w, conv_b, bn_g, bn_b, bn_m, bn_v,
                                        res_w, res_b, res_g, res_be, res_m, res_v, ws);
    gcn_fused_kernel<<<NB * (TDIM / TT), 256, 0, stream>>>(x, ws, out);
}